// SetAttentionField_71837622992976
// MI455X (gfx1250) — compile-verified
//
#include <hip/hip_runtime.h>
#include <math.h>

typedef __attribute__((ext_vector_type(2))) float v2f;
typedef __attribute__((ext_vector_type(8))) float v8f;

#define NROWS 512   // rows in feat_x / feat_pos / feat_neg
#define DIN   512
#define DHID  256
#define DREP  256
#define DSH   128   // score hidden

__device__ __forceinline__ float gelu_exact(float x) {
    // torch nn.GELU default: exact erf form
    return 0.5f * x * (1.0f + erff(x * 0.70710678118654752440f));
}

// ---------------------------------------------------------------------------
// WMMA f32 GEMM: C(MxN) = epi(A(MxK) @ B)
//   TRANSB=false: B stored (K x N) row-major
//   TRANSB=true : B stored (N x K) row-major (logical B^T), used for Gram
//   EPI: 0 = none, 1 = +bias[n], 2 = gelu(acc + bias[n])
//
// One wave computes a 16(M) x 64(N) strip = 4 accumulators that share one
// A fragment per K-step (4 V_WMMA_F32_16X16X4_F32 per A load -> 4x the
// compute intensity of one-tile-per-wave). All inner-loop addressing is
// pointer-increment form to kill per-iteration SALU/VALU address math.
// Block = 128 threads = 4 waves stacked along M. Requires M%64==0, N%64==0,
// K%4==0 (all shapes here satisfy this).
//
// A 16x4 f32 operand: lane&15 = M row, (lane>>4)*2 = K-pair base, the two
// VGPRs hold consecutive K. B 4x16 mirrors it with N in lane&15.
// C/D: vgpr v, lane l -> C[M = v + 8*(l>>4)][N = l&15].
// ---------------------------------------------------------------------------
template<int EPI, bool TRANSB>
__global__ void gemm_wmma_f32(const float* __restrict__ A,
                              const float* __restrict__ B,
                              const float* __restrict__ bias,
                              float* __restrict__ C,
                              int M, int N, int K) {
    const int lane = threadIdx.x & 31;
    const int wave = threadIdx.x >> 5;
    const int tn = blockIdx.x;               // 64-wide tile column
    const int tm = blockIdx.y * 4 + wave;    // 16-tall tile row (per wave)
    const int r  = lane & 15;
    const int kg = (lane >> 4) * 2;          // K-pair base for this half-wave

    const int arow = tm * 16 + r;
    v8f acc0 = {}, acc1 = {}, acc2 = {}, acc3 = {};

    const float* ap = A + arow * K + kg;
    if (TRANSB) {
        const float* bp = B + (tn * 64 + r) * K + kg;   // tiles offset by 16*K
        for (int k0 = 0; k0 < K; k0 += 4) {
            v2f a = { ap[0], ap[1] };
            v2f b0 = { bp[0 * 16 * K], bp[0 * 16 * K + 1] };
            v2f b1 = { bp[1 * 16 * K], bp[1 * 16 * K + 1] };
            v2f b2 = { bp[2 * 16 * K], bp[2 * 16 * K + 1] };
            v2f b3 = { bp[3 * 16 * K], bp[3 * 16 * K + 1] };
            acc0 = __builtin_amdgcn_wmma_f32_16x16x4_f32(false, a, false, b0, (short)0, acc0, false, false);
            acc1 = __builtin_amdgcn_wmma_f32_16x16x4_f32(false, a, false, b1, (short)0, acc1, false, false);
            acc2 = __builtin_amdgcn_wmma_f32_16x16x4_f32(false, a, false, b2, (short)0, acc2, false, false);
            acc3 = __builtin_amdgcn_wmma_f32_16x16x4_f32(false, a, false, b3, (short)0, acc3, false, false);
            ap += 4;
            bp += 4;
        }
    } else {
        const float* bp = B + kg * N + tn * 64 + r;     // tiles offset by 16 cols
        for (int k0 = 0; k0 < K; k0 += 4) {
            v2f a = { ap[0], ap[1] };
            v2f b0 = { bp[0 * 16], bp[N + 0 * 16] };
            v2f b1 = { bp[1 * 16], bp[N + 1 * 16] };
            v2f b2 = { bp[2 * 16], bp[N + 2 * 16] };
            v2f b3 = { bp[3 * 16], bp[N + 3 * 16] };
            acc0 = __builtin_amdgcn_wmma_f32_16x16x4_f32(false, a, false, b0, (short)0, acc0, false, false);
            acc1 = __builtin_amdgcn_wmma_f32_16x16x4_f32(false, a, false, b1, (short)0, acc1, false, false);
            acc2 = __builtin_amdgcn_wmma_f32_16x16x4_f32(false, a, false, b2, (short)0, acc2, false, false);
            acc3 = __builtin_amdgcn_wmma_f32_16x16x4_f32(false, a, false, b3, (short)0, acc3, false, false);
            ap += 4;
            bp += 4 * N;
        }
    }

    const int m0 = tm * 16 + 8 * (lane >> 4);
    v8f accs[4] = { acc0, acc1, acc2, acc3 };
#pragma unroll
    for (int t = 0; t < 4; ++t) {
        const int cn = tn * 64 + t * 16 + r;
        float bv = (EPI >= 1) ? bias[cn] : 0.0f;
#pragma unroll
        for (int v = 0; v < 8; ++v) {
            float val = accs[t][v] + bv;
            if (EPI == 2) val = gelu_exact(val);
            C[(m0 + v) * N + cn] = val;
        }
    }
}

// ---------------------------------------------------------------------------
// WA = W1x - W1d ; WB = W1y + W1d   (each 256x128, from s_w1 (769x128))
// ---------------------------------------------------------------------------
__global__ void combine_score_w(const float* __restrict__ s_w1,
                                float* __restrict__ WA,
                                float* __restrict__ WB) {
    int i = blockIdx.x * blockDim.x + threadIdx.x;
    if (i < DREP * DSH) {
        float wx = s_w1[i];
        float wy = s_w1[DREP * DSH + i];
        float wd = s_w1[2 * DREP * DSH + i];
        WA[i] = wx - wd;
        WB[i] = wy + wd;
    }
}

// ---------------------------------------------------------------------------
// out[n] = || R[n,:] ||^2   (one wave per row, K=256)
// ---------------------------------------------------------------------------
__global__ void row_sqnorm(const float* __restrict__ R, float* __restrict__ out) {
    int n = blockIdx.x;
    int lane = threadIdx.x;
    float s = 0.0f;
    for (int k = lane; k < DREP; k += 32) {
        float v = R[n * DREP + k];
        s += v * v;
    }
#pragma unroll
    for (int off = 16; off > 0; off >>= 1) s += __shfl_xor(s, off, 32);
    if (lane == 0) out[n] = s;
}

// ---------------------------------------------------------------------------
// Pairwise logits:
//   L[n,m] = sum_k gelu(A[n,k] + B[m,k] + norm(n,m)*w1n[k]) * w2[k] + b2
//   norm = sqrt(max(xs[n] + ys[m] - 2*G[n,m], 0))
// grid = 512 (one block per n), block = 256 (8 waves, wave-per-m).
// Each lane owns 4 CONTIGUOUS k (k = 4*lane + c) so the per-m B-row read is
// one global_load_b128 per lane; A-row / w1n / s_w2 staged in LDS.
// ---------------------------------------------------------------------------
__global__ void pair_logits(const float* __restrict__ Arow,  // 512x128
                            const float* __restrict__ Brow,  // 512x128
                            const float* __restrict__ xs,    // 512
                            const float* __restrict__ ys,    // 512
                            const float* __restrict__ G,     // 512x512
                            const float* __restrict__ w1n,   // 128
                            const float* __restrict__ w2,    // 128
                            const float* __restrict__ b2,    // 1
                            float* __restrict__ L) {         // 512x512
    __shared__ float sA[DSH], sN[DSH], sW[DSH];
    const int n = blockIdx.x;
    const int t = threadIdx.x;
    if (t < DSH) {
        sA[t] = Arow[n * DSH + t];
        sN[t] = w1n[t];
        sW[t] = w2[t];
    }
    __syncthreads();
    const int lane = t & 31;
    const int wave = t >> 5;
    const float xsn = xs[n];
    const float bias2 = b2[0];
    const float4* Bv = (const float4*)Brow;          // 512 x 32 float4
    for (int m = wave; m < NROWS; m += 8) {
        float d2 = fmaxf(xsn + ys[m] - 2.0f * G[n * NROWS + m], 0.0f);
        float nrm = sqrtf(d2);
        float4 bq = Bv[m * 32 + lane];               // k = 4*lane .. 4*lane+3
        const int kb = 4 * lane;
        float acc;
        acc  = gelu_exact(sA[kb + 0] + bq.x + nrm * sN[kb + 0]) * sW[kb + 0];
        acc += gelu_exact(sA[kb + 1] + bq.y + nrm * sN[kb + 1]) * sW[kb + 1];
        acc += gelu_exact(sA[kb + 2] + bq.z + nrm * sN[kb + 2]) * sW[kb + 2];
        acc += gelu_exact(sA[kb + 3] + bq.w + nrm * sN[kb + 3]) * sW[kb + 3];
#pragma unroll
        for (int off = 16; off > 0; off >>= 1) acc += __shfl_down(acc, off, 32);
        if (lane == 0) L[n * NROWS + m] = acc + bias2;
    }
}

// ---------------------------------------------------------------------------
// In-place softmax along each 512-wide row; one wave per row.
// ---------------------------------------------------------------------------
__global__ void softmax_rows(float* __restrict__ L) {
    int n = blockIdx.x;
    int lane = threadIdx.x;
    float vmax = -INFINITY;
    for (int m = lane; m < NROWS; m += 32) vmax = fmaxf(vmax, L[n * NROWS + m]);
#pragma unroll
    for (int off = 16; off > 0; off >>= 1) vmax = fmaxf(vmax, __shfl_xor(vmax, off, 32));
    float sum = 0.0f;
    for (int m = lane; m < NROWS; m += 32) {
        float e = __expf(L[n * NROWS + m] - vmax);
        L[n * NROWS + m] = e;
        sum += e;
    }
#pragma unroll
    for (int off = 16; off > 0; off >>= 1) sum += __shfl_xor(sum, off, 32);
    float inv = 1.0f / sum;
    for (int m = lane; m < NROWS; m += 32) L[n * NROWS + m] *= inv;
}

// ---------------------------------------------------------------------------
// D = P - Q (elementwise); x_rep terms cancel in s_pos - s_neg.
// ---------------------------------------------------------------------------
__global__ void ew_sub(const float* __restrict__ P, const float* __restrict__ Q,
                       float* __restrict__ D, int n) {
    int i = blockIdx.x * blockDim.x + threadIdx.x;
    if (i < n) D[i] = P[i] - Q[i];
}

// ---------------------------------------------------------------------------
extern "C" void kernel_launch(void* const* d_in, const int* in_sizes, int n_in,
                              void* d_out, int out_size, void* d_ws, size_t ws_size,
                              hipStream_t stream) {
    const float* feat_x  = (const float*)d_in[0];
    const float* feat_p  = (const float*)d_in[1];
    const float* feat_n  = (const float*)d_in[2];
    const float* g_w1    = (const float*)d_in[3];
    const float* g_b1    = (const float*)d_in[4];
    const float* g_w2    = (const float*)d_in[5];
    const float* g_b2    = (const float*)d_in[6];
    const float* out_w   = (const float*)d_in[7];
    const float* s_w1    = (const float*)d_in[8];
    const float* s_b1    = (const float*)d_in[9];
    const float* s_w2    = (const float*)d_in[10];
    const float* s_b2    = (const float*)d_in[11];
    float* out = (float*)d_out;

    float* w = (float*)d_ws;
    float* rep_x = w; w += NROWS * DREP;
    float* rep_p = w; w += NROWS * DREP;
    float* rep_n = w; w += NROWS * DREP;
    float* Htmp  = w; w += NROWS * DHID;
    float* WA    = w; w += DREP * DSH;
    float* WB    = w; w += DREP * DSH;
    float* Ax    = w; w += NROWS * DSH;
    float* Bp    = w; w += NROWS * DSH;
    float* Bn    = w; w += NROWS * DSH;
    float* xs    = w; w += NROWS;
    float* ysp   = w; w += NROWS;
    float* ysn   = w; w += NROWS;
    float* Gp    = w; w += NROWS * NROWS;
    float* Gn    = w; w += NROWS * NROWS;
    float* Up    = w; w += NROWS * DREP;
    float* Un    = w; w += NROWS * DREP;

    dim3 blk(128);
    // ---- g MLP: rep = gelu(F @ g_w1 + b1) @ g_w2 + b2 ----
    {
        dim3 g1(DHID / 64, NROWS / 64), g2(DREP / 64, NROWS / 64);
        gemm_wmma_f32<2, false><<<g1, blk, 0, stream>>>(feat_x, g_w1, g_b1, Htmp, NROWS, DHID, DIN);
        gemm_wmma_f32<1, false><<<g2, blk, 0, stream>>>(Htmp, g_w2, g_b2, rep_x, NROWS, DREP, DHID);
        gemm_wmma_f32<2, false><<<g1, blk, 0, stream>>>(feat_p, g_w1, g_b1, Htmp, NROWS, DHID, DIN);
        gemm_wmma_f32<1, false><<<g2, blk, 0, stream>>>(Htmp, g_w2, g_b2, rep_p, NROWS, DREP, DHID);
        gemm_wmma_f32<2, false><<<g1, blk, 0, stream>>>(feat_n, g_w1, g_b1, Htmp, NROWS, DHID, DIN);
        gemm_wmma_f32<1, false><<<g2, blk, 0, stream>>>(Htmp, g_w2, g_b2, rep_n, NROWS, DREP, DHID);
    }

    // ---- score-weight decomposition ----
    combine_score_w<<<(DREP * DSH + 255) / 256, 256, 0, stream>>>(s_w1, WA, WB);

    // ---- A = x_rep@(W1x-W1d), B = y_rep@(W1y+W1d) + b1 ----
    {
        dim3 g(DSH / 64, NROWS / 64);
        gemm_wmma_f32<0, false><<<g, blk, 0, stream>>>(rep_x, WA, nullptr, Ax, NROWS, DSH, DREP);
        gemm_wmma_f32<1, false><<<g, blk, 0, stream>>>(rep_p, WB, s_b1, Bp, NROWS, DSH, DREP);
        gemm_wmma_f32<1, false><<<g, blk, 0, stream>>>(rep_n, WB, s_b1, Bn, NROWS, DSH, DREP);
    }

    // ---- squared norms + Gram matrices for ||y - x|| ----
    row_sqnorm<<<NROWS, 32, 0, stream>>>(rep_x, xs);
    row_sqnorm<<<NROWS, 32, 0, stream>>>(rep_p, ysp);
    row_sqnorm<<<NROWS, 32, 0, stream>>>(rep_n, ysn);
    {
        dim3 g(NROWS / 64, NROWS / 64);
        gemm_wmma_f32<0, true><<<g, blk, 0, stream>>>(rep_x, rep_p, nullptr, Gp, NROWS, NROWS, DREP);
        gemm_wmma_f32<0, true><<<g, blk, 0, stream>>>(rep_x, rep_n, nullptr, Gn, NROWS, NROWS, DREP);
    }

    // ---- pairwise logits + softmax (in-place -> attention weights) ----
    const float* w1n = s_w1 + 3 * DREP * DSH;   // row 768 of s_w1
    pair_logits<<<NROWS, 256, 0, stream>>>(Ax, Bp, xs, ysp, Gp, w1n, s_w2, s_b2, Gp);
    pair_logits<<<NROWS, 256, 0, stream>>>(Ax, Bn, xs, ysn, Gn, w1n, s_w2, s_b2, Gn);
    softmax_rows<<<NROWS, 32, 0, stream>>>(Gp);
    softmax_rows<<<NROWS, 32, 0, stream>>>(Gn);

    // ---- U = W @ y_rep ; x_rep cancels in (pos - neg) ----
    {
        dim3 g(DREP / 64, NROWS / 64);
        gemm_wmma_f32<0, false><<<g, blk, 0, stream>>>(Gp, rep_p, nullptr, Up, NROWS, DREP, NROWS);
        gemm_wmma_f32<0, false><<<g, blk, 0, stream>>>(Gn, rep_n, nullptr, Un, NROWS, DREP, NROWS);
    }
    ew_sub<<<(NROWS * DREP + 255) / 256, 256, 0, stream>>>(Up, Un, Up, NROWS * DREP);

    // ---- final projection: out = (Up - Un) @ out_w ----
    {
        dim3 g(DIN / 64, NROWS / 64);
        gemm_wmma_f32<0, false><<<g, blk, 0, stream>>>(Up, out_w, nullptr, out, NROWS, DIN, DREP);
    }
}